// thetaRNNLayer_17042430231265
// MI455X (gfx1250) — compile-verified
//
#include <hip/hip_runtime.h>
#include <cmath>

typedef __attribute__((ext_vector_type(2))) float v2f;
typedef __attribute__((ext_vector_type(8))) float v8f;
typedef __attribute__((ext_vector_type(4))) int   v4i;

#define SEQ 4096
#define HID 512
#define NTH 5
#define SH (SEQ * HID)

// ---------------------------------------------------------------------------
// Async global->LDS copy (gfx1250). Builtin when available, else inline asm.
// Builtin signature (from hipcc diagnostic): param0 is a typed global-AS
// vector pointer, so cast to address_space(1)/(3) v4i*.
// ---------------------------------------------------------------------------
#if __has_builtin(__builtin_amdgcn_global_load_async_to_lds_b128)
__device__ __forceinline__ void async_cp_b128(const float* gsrc, float* ldst) {
  __builtin_amdgcn_global_load_async_to_lds_b128(
      (__attribute__((address_space(1))) v4i*)gsrc,
      (__attribute__((address_space(3))) v4i*)ldst, 0, 0);
}
#else
__device__ __forceinline__ void async_cp_b128(const float* gsrc, float* ldst) {
  unsigned loff = (unsigned)(unsigned long long)ldst;
  asm volatile("global_load_async_to_lds_b128 %0, %1, off"
               :: "v"(loff), "v"(gsrc) : "memory");
}
#endif

__device__ __forceinline__ void wait_async0() {
#if __has_builtin(__builtin_amdgcn_s_wait_asynccnt)
  __builtin_amdgcn_s_wait_asynccnt(0);
#else
  asm volatile("s_wait_asynccnt 0x0" ::: "memory");
#endif
}

// ---------------------------------------------------------------------------
// Init: zero the barrier counter, seed the ping-pong h buffer with `state`.
// ---------------------------------------------------------------------------
__global__ void init_kernel(const float* __restrict__ state,
                            float* __restrict__ hbuf,
                            unsigned* __restrict__ cnt) {
  const int t = threadIdx.x;
  if (t == 0) *cnt = 0u;
  hbuf[t]       = state[t];  // buffer 0 = h0
  hbuf[HID + t] = 0.f;
}

// ---------------------------------------------------------------------------
// Fused GEMM:  C[s,n] = ep( A[s,:] . W[n,:] + bias[n] + add[s,n] )
// V_WMMA_F32_16X16X4_F32, 64x64 block tile, 8 waves (16 rows x 32 cols each),
// double-buffered LDS filled by async global->LDS copies (ASYNCcnt pipeline).
// ---------------------------------------------------------------------------
__launch_bounds__(256)
__global__ void gemm_wmma(const float* __restrict__ A,    // [4096,512]
                          const float* __restrict__ W,    // [512,512] row-major (n,k)
                          const float* __restrict__ bias, // [512]
                          const float* __restrict__ add,  // [4096,512]
                          float* __restrict__ C,          // [4096,512]
                          int do_tanh) {
  constexpr int LDT = 36;                  // padded LDS row stride (floats)
  __shared__ __align__(16) float As[2][64 * LDT];
  __shared__ __align__(16) float Bs[2][64 * LDT];

  const int tid  = threadIdx.x;
  const int wave = tid >> 5;
  const int lane = tid & 31;
  const int n0   = blockIdx.x * 64;       // output-col tile
  const int s0   = blockIdx.y * 64;       // output-row tile
  const int mrow = (wave & 3) * 16;
  const int ncol = (wave >> 2) * 32;
  const int r    = lane & 15;
  const int hi   = lane >> 4;

  v8f acc0 = {};
  v8f acc1 = {};

  // Issue one 64x32 A tile + 64x32 B tile into LDS buffer `buf` (async).
  auto issue_chunk = [&](int k0, int buf) {
#pragma unroll
    for (int it = 0; it < 2; ++it) {
      const int idx = tid + it * 256;
      const int row = idx >> 3;
      const int k4  = (idx & 7) * 4;
      async_cp_b128(&A[(s0 + row) * 512 + k0 + k4], &As[buf][row * LDT + k4]);
      async_cp_b128(&W[(n0 + row) * 512 + k0 + k4], &Bs[buf][row * LDT + k4]);
    }
  };

  issue_chunk(0, 0);                       // prime the pipeline

  for (int i = 0; i < 16; ++i) {           // 16 k-chunks of 32
    const int buf = i & 1;
    wait_async0();                         // this wave's copies for chunk i done
    __syncthreads();                       // all waves' copies visible; other
                                           // buffer no longer being read
    if (i + 1 < 16) issue_chunk((i + 1) * 32, buf ^ 1);

#pragma unroll
    for (int kk = 0; kk < 32; kk += 4) {
      // A 16x4 layout: lanes 0-15 -> K={kk,kk+1}, lanes 16-31 -> K={kk+2,kk+3}
      v2f a, b0, b1;
      a.x  = As[buf][(mrow + r) * LDT + kk + 2 * hi];
      a.y  = As[buf][(mrow + r) * LDT + kk + 2 * hi + 1];
      // B 4x16: B[k][n] = W[n][k]; lanes hold n, VGPR pair holds K like A
      b0.x = Bs[buf][(ncol + r) * LDT + kk + 2 * hi];
      b0.y = Bs[buf][(ncol + r) * LDT + kk + 2 * hi + 1];
      b1.x = Bs[buf][(ncol + 16 + r) * LDT + kk + 2 * hi];
      b1.y = Bs[buf][(ncol + 16 + r) * LDT + kk + 2 * hi + 1];
      acc0 = __builtin_amdgcn_wmma_f32_16x16x4_f32(false, a, false, b0,
                                                   (short)0, acc0, false, false);
      acc1 = __builtin_amdgcn_wmma_f32_16x16x4_f32(false, a, false, b1,
                                                   (short)0, acc1, false, false);
    }
  }

  // Epilogue. C/D layout: VGPR v -> row v (lanes<16) or v+8 (lanes>=16), lane&15 = col.
  const int c   = lane & 15;
  const int n_a = n0 + ncol + c;
  const int n_b = n0 + ncol + 16 + c;
  const float ba = bias[n_a];
  const float bb = bias[n_b];
#pragma unroll
  for (int v = 0; v < 8; ++v) {
    const int s = s0 + mrow + v + 8 * hi;
    float x0 = acc0[v] + ba + add[s * 512 + n_a];
    float x1 = acc1[v] + bb + add[s * 512 + n_b];
    if (do_tanh) { x0 = tanhf(x0); x1 = tanhf(x1); }
    C[s * 512 + n_a] = x0;
    C[s * 512 + n_b] = x1;
  }
}

// ---------------------------------------------------------------------------
// Serial scan: 4 workgroups x 1024 threads. WG w owns W_hh rows [w*128, +128)
// resident in LDS (256 KB fp32, fits CDNA5's 320 KB WGP LDS). Per step:
// stage h in LDS, 8 threads/row interleaved dot + shfl_xor reduce, tanh,
// ping-pong h via global, device-scope atomic barrier.
// ---------------------------------------------------------------------------
__launch_bounds__(1024)
__global__ void scan_kernel(const float* __restrict__ preX, // x@Wih^T + b + internal[0]
                            const float* __restrict__ Whh,  // [512,512]
                            float* __restrict__ hbuf,       // [2,512] ping-pong
                            unsigned* __restrict__ cnt,
                            float* __restrict__ out) {      // d_out base
  __shared__ __align__(16) float Wl[128 * 512];
  __shared__ __align__(16) float hl[512];

  const int tid = threadIdx.x;
  const int wg  = blockIdx.x;           // 0..3
  const int row = tid >> 3;             // 0..127
  const int sub = tid & 7;

  const float4* wsrc = (const float4*)(Whh + wg * 128 * 512);
  for (int i = tid; i < 128 * 512 / 4; i += 1024) ((float4*)Wl)[i] = wsrc[i];
  __syncthreads();

  const int ig = wg * 128 + row;        // global output row

  for (int t = 0; t < SEQ; ++t) {
    const float* hcur = hbuf + ((t & 1) * 512);
    float*       hnxt = hbuf + (((t + 1) & 1) * 512);

    if (tid < 512) hl[tid] = hcur[tid];
    __syncthreads();

    float sum = 0.f;
#pragma unroll
    for (int m = 0; m < 16; ++m) {      // interleaved k = sub*4 + m*32 (bank spread)
      const int k  = sub * 4 + m * 32;
      const float4 wv = *(const float4*)&Wl[row * 512 + k];
      const float4 hv = *(const float4*)&hl[k];
      sum += wv.x * hv.x + wv.y * hv.y + wv.z * hv.z + wv.w * hv.w;
    }
    sum += __shfl_xor(sum, 1);
    sum += __shfl_xor(sum, 2);
    sum += __shfl_xor(sum, 4);

    if (sub == 0) {
      const float val = tanhf(preX[t * 512 + ig] + sum);
      out[t * 512 + ig] = val;          // outputs[0][t]
      hnxt[ig] = val;
      if (t == SEQ - 1) out[6 * SH + ig] = val;   // final_state
    }

    __threadfence();
    __syncthreads();
    if (tid == 0) {
      __hip_atomic_fetch_add(cnt, 1u, __ATOMIC_RELEASE, __HIP_MEMORY_SCOPE_AGENT);
      while (__hip_atomic_load(cnt, __ATOMIC_ACQUIRE, __HIP_MEMORY_SCOPE_AGENT) <
             4u * (unsigned)(t + 1))
        __builtin_amdgcn_s_sleep(1);
    }
    __syncthreads();
  }
}

// ---------------------------------------------------------------------------
extern "C" void kernel_launch(void* const* d_in, const int* in_sizes, int n_in,
                              void* d_out, int out_size, void* d_ws, size_t ws_size,
                              hipStream_t stream) {
  (void)in_sizes; (void)n_in; (void)out_size; (void)ws_size;
  const float* x        = (const float*)d_in[0];  // [1,4096,512]
  const float* internal = (const float*)d_in[1];  // [6,4096,512]
  const float* state    = (const float*)d_in[2];  // [1,1,512]
  const float* W_ih     = (const float*)d_in[3];  // [512,512]
  const float* W_hh     = (const float*)d_in[4];  // [512,512]
  const float* b        = (const float*)d_in[5];  // [512]

  float* out  = (float*)d_out;
  float* preX = (float*)d_ws;                     // [4096,512]  (8 MB)
  float* hbuf = preX + SH;                        // [2,512]
  unsigned* cnt = (unsigned*)(hbuf + 2 * HID);    // barrier counter

  // 1) reset barrier + seed h0
  init_kernel<<<1, HID, 0, stream>>>(state, hbuf, cnt);

  // 2) preX = x @ W_ih^T + b + internal[0]   (fold noise+bias into scan drive)
  gemm_wmma<<<dim3(8, 64), 256, 0, stream>>>(x, W_ih, b, internal, preX, 0);

  // 3) serial recurrence -> outputs[0] + final_state
  scan_kernel<<<4, 1024, 0, stream>>>(preX, W_hh, hbuf, cnt, out);

  // 4) theta rollouts: outputs[j] = tanh(b + outputs[j-1] @ W_hh^T + internal[j])
  for (int j = 1; j <= NTH; ++j)
    gemm_wmma<<<dim3(8, 64), 256, 0, stream>>>(out + (j - 1) * SH, W_hh, b,
                                               internal + j * SH, out + j * SH, 1);
}